// GIBGAT_13134009991727
// MI455X (gfx1250) — compile-verified
//
#include <hip/hip_runtime.h>
#include <hip/hip_bf16.h>

typedef __attribute__((ext_vector_type(2))) float v2f;
typedef __attribute__((ext_vector_type(8))) float v8f;

#define F_IN 128
#define C1   64
#define C2   16
#define NEG_SLOPE 0.2f

// ---------------------------------------------------------------------------
// GEMM1: h1[N,64] = x[N,128] @ W1[128,64]   (fp32 WMMA 16x16x4)
// W1 staged in LDS pre-packed in WMMA-B fragment order:
//   element W[k][c] -> sW[q*256 + c*4 + j],  q=k/4, j=k%4
// so each lane's B fragment is ONE aligned ds_load_b64 (conflict-free banks).
// ---------------------------------------------------------------------------
__global__ void __launch_bounds__(256)
gemm1_wmma(const float* __restrict__ x, const float* __restrict__ W1,
           float* __restrict__ h1, int Nn) {
  __shared__ float sW[F_IN * C1];                        // 32 KB (packed)
  for (int i = threadIdx.x; i < (F_IN * C1) / 4; i += 256) {
    float4 v = ((const float4*)W1)[i];
    const int k  = i >> 4;            // row of W1 (64 cols = 16 float4s)
    const int c0 = (i & 15) * 4;
    float* dst = sW + (k >> 2) * 256 + (k & 3);          // +c*4 below
    dst[(c0 + 0) * 4] = v.x;
    dst[(c0 + 1) * 4] = v.y;
    dst[(c0 + 2) * 4] = v.z;
    dst[(c0 + 3) * 4] = v.w;
  }
  __syncthreads();

  const int wave = threadIdx.x >> 5;
  const int lane = threadIdx.x & 31;
  const int tile = blockIdx.x * 8 + wave;
  if (tile * 16 >= Nn) return;                           // wave-uniform
  const int row0 = tile * 16;
  const int r    = lane & 15;                            // row (A) / col (B,D)
  const int kh   = (lane >> 4) << 1;                     // K sub-offset 0 or 2
  const int arow = min(row0 + r, Nn - 1);                // clamp tail loads

  v8f acc0 = {}, acc1 = {}, acc2 = {}, acc3 = {};
  const float* ap = x + (size_t)arow * F_IN + kh;
  const float* bbase = sW + r * 4 + kh;                  // + q*256 + t*64
  for (int k0 = 0; k0 < F_IN; k0 += 4) {
    v2f a; a.x = ap[k0]; a.y = ap[k0 + 1];
    const float* bq = bbase + (k0 >> 2) * 256;
    v2f b0 = *(const v2f*)(bq);
    v2f b1 = *(const v2f*)(bq + 64);
    v2f b2 = *(const v2f*)(bq + 128);
    v2f b3 = *(const v2f*)(bq + 192);
    acc0 = __builtin_amdgcn_wmma_f32_16x16x4_f32(false, a, false, b0, (short)0, acc0, false, false);
    acc1 = __builtin_amdgcn_wmma_f32_16x16x4_f32(false, a, false, b1, (short)0, acc1, false, false);
    acc2 = __builtin_amdgcn_wmma_f32_16x16x4_f32(false, a, false, b2, (short)0, acc2, false, false);
    acc3 = __builtin_amdgcn_wmma_f32_16x16x4_f32(false, a, false, b3, (short)0, acc3, false, false);
  }

  const int mbase = (lane >> 4) * 8;                     // D layout: M = i (+8)
  for (int i = 0; i < 8; ++i) {
    const int row = row0 + mbase + i;
    if (row < Nn) {
      float* op = h1 + (size_t)row * C1 + r;
      op[0]  = acc0[i];
      op[16] = acc1[i];
      op[32] = acc2[i];
      op[48] = acc3[i];
    }
  }
}

// ---------------------------------------------------------------------------
// GEMM2: h2[N,16] = h[N,64] @ W2[64,16]   (fp32 WMMA 16x16x4)
// same packed-B LDS scheme: W2[k][c] -> sW[q*64 + c*4 + j]
// ---------------------------------------------------------------------------
__global__ void __launch_bounds__(256)
gemm2_wmma(const float* __restrict__ h, const float* __restrict__ W2,
           float* __restrict__ h2, int Nn) {
  __shared__ float sW[C1 * C2];                          // 4 KB (packed)
  for (int i = threadIdx.x; i < (C1 * C2) / 4; i += 256) {
    float4 v = ((const float4*)W2)[i];
    const int k  = i >> 2;            // row of W2 (16 cols = 4 float4s)
    const int c0 = (i & 3) * 4;
    float* dst = sW + (k >> 2) * 64 + (k & 3);
    dst[(c0 + 0) * 4] = v.x;
    dst[(c0 + 1) * 4] = v.y;
    dst[(c0 + 2) * 4] = v.z;
    dst[(c0 + 3) * 4] = v.w;
  }
  __syncthreads();

  const int wave = threadIdx.x >> 5;
  const int lane = threadIdx.x & 31;
  const int tile = blockIdx.x * 8 + wave;
  if (tile * 16 >= Nn) return;
  const int row0 = tile * 16;
  const int r    = lane & 15;
  const int kh   = (lane >> 4) << 1;
  const int arow = min(row0 + r, Nn - 1);

  v8f acc = {};
  const float* ap = h + (size_t)arow * C1 + kh;
  const float* bbase = sW + r * 4 + kh;
  for (int k0 = 0; k0 < C1; k0 += 4) {
    v2f a; a.x = ap[k0]; a.y = ap[k0 + 1];
    v2f b = *(const v2f*)(bbase + (k0 >> 2) * 64);
    acc = __builtin_amdgcn_wmma_f32_16x16x4_f32(false, a, false, b, (short)0, acc, false, false);
  }

  const int mbase = (lane >> 4) * 8;
  for (int i = 0; i < 8; ++i) {
    const int row = row0 + mbase + i;
    if (row < Nn) h2[(size_t)row * C2 + r] = acc[i];
  }
}

// ---------------------------------------------------------------------------
// per-node attention dots: s[n] = h[n]·a_src, d[n] = h[n]·a_dst
// ---------------------------------------------------------------------------
template <int C>
__global__ void node_dots(const float* __restrict__ h,
                          const float* __restrict__ asrc,
                          const float* __restrict__ adst,
                          float* __restrict__ s, float* __restrict__ d, int Nn) {
  int n = blockIdx.x * blockDim.x + threadIdx.x;
  if (n >= Nn) return;
  const float* hp = h + (size_t)n * C;
  float ss = 0.f, dd = 0.f;
  for (int f = 0; f < C; ++f) { float v = hp[f]; ss += v * asrc[f]; dd += v * adst[f]; }
  s[n] = ss; d[n] = dd;
}

// init accumulators: out_acc = 0, amax = INT_MIN (key of -inf), denom = 0
__global__ void init_accum(float* __restrict__ outacc, size_t acc_elems,
                           int* __restrict__ amax, float* __restrict__ denom, int Nn) {
  size_t i = (size_t)blockIdx.x * blockDim.x + threadIdx.x;
  if (i < acc_elems) outacc[i] = 0.f;
  if (i < (size_t)Nn) { amax[i] = (int)0x80000000; denom[i] = 0.f; }
}

__device__ __forceinline__ int fkey(float f) {
  int k = __float_as_int(f);
  return k >= 0 ? k : (k ^ 0x7fffffff);
}
__device__ __forceinline__ float fkey_inv(int k) {
  return __int_as_float(k >= 0 ? k : (k ^ 0x7fffffff));
}
__device__ __forceinline__ void edge_sd(const int* ei, int E, int e, int& s, int& d) {
  if (e < E) { s = ei[e]; d = ei[E + e]; } else { s = d = e - E; }   // self loops
}

// pass 1: alpha = leaky_relu(s[src]+d[dst]); segment-max over dst (int-key atomicMax)
__global__ void edge_max_kernel(const int* __restrict__ ei, int E, int Etot,
                                const float* __restrict__ s, const float* __restrict__ d,
                                float* __restrict__ alpha_buf, int* __restrict__ amax) {
  int e = blockIdx.x * blockDim.x + threadIdx.x;
  if (e >= Etot) return;
  int sn, dn; edge_sd(ei, E, e, sn, dn);
  float a = s[sn] + d[dn];
  a = a > 0.f ? a : NEG_SLOPE * a;
  alpha_buf[e] = a;
  atomicMax(amax + dn, fkey(a));
}

// pass 2: e_w = exp(alpha - amax[dst]); segment-sum of e_w
__global__ void edge_sum_kernel(const int* __restrict__ ei, int E, int Etot,
                                float* __restrict__ alpha_buf,
                                const int* __restrict__ amax, float* __restrict__ denom) {
  int e = blockIdx.x * blockDim.x + threadIdx.x;
  if (e >= Etot) return;
  int sn, dn; edge_sd(ei, E, e, sn, dn);
  float ew = __expf(alpha_buf[e] - fkey_inv(amax[dn]));
  alpha_buf[e] = ew;                                     // overwrite with e_w
  atomicAdd(denom + dn, ew);
}

// pass 3: out[dst] += h[src] * e_w / (denom[dst]+1e-16); 4 features per thread
template <int C>
__global__ void edge_agg_kernel(const int* __restrict__ ei, int E, int Etot,
                                const float* __restrict__ ewb, const float* __restrict__ denom,
                                const float* __restrict__ hsrc, float* __restrict__ outacc) {
  const int chunks = C / 4;
  long long tid = (long long)blockIdx.x * blockDim.x + threadIdx.x;
  int e  = (int)(tid / chunks);
  int fo = (int)(tid % chunks) * 4;
  if (e >= Etot) return;
  int sn, dn; edge_sd(ei, E, e, sn, dn);
  float w = ewb[e] / (denom[dn] + 1e-16f);
  float4 hv = *(const float4*)(hsrc + (size_t)sn * C + fo);
  float* op = outacc + (size_t)dn * C + fo;
  atomicAdd(op + 0, hv.x * w);
  atomicAdd(op + 1, hv.y * w);
  atomicAdd(op + 2, hv.z * w);
  atomicAdd(op + 3, hv.w * w);
}

__device__ __forceinline__ float softplusf(float v) {
  return v > 0.f ? v + log1pf(__expf(-v)) : log1pf(__expf(v));
}

// finish layer 1: out1 = acc + b1; h = elu(out1); ixz1 = KL(mean=out1[:32], std=softplus(out1[32:]))
__global__ void finish1_kernel(const float* __restrict__ acc, const float* __restrict__ b1,
                               float* __restrict__ hbuf, float* __restrict__ ixz1, int Nn) {
  size_t tid = (size_t)blockIdx.x * blockDim.x + threadIdx.x;
  int n = (int)(tid >> 6), f = (int)(tid & 63);
  if (n >= Nn) return;
  float v = acc[(size_t)n * C1 + f] + b1[f];
  hbuf[(size_t)n * C1 + f] = v > 0.f ? v : (__expf(v) - 1.f);
  if (f < 32) {
    float sp  = acc[(size_t)n * C1 + 32 + f] + b1[32 + f];
    float sd  = softplusf(sp) + 1e-10f;
    ixz1[(size_t)n * 32 + f] = -__logf(sd) + 0.5f * (sd * sd + v * v - 1.f);
  }
}

// finish layer 2: out2 = acc + b2 -> d_out; ixz2 likewise
__global__ void finish2_kernel(const float* __restrict__ acc, const float* __restrict__ b2,
                               float* __restrict__ out2, float* __restrict__ ixz2, int Nn) {
  size_t tid = (size_t)blockIdx.x * blockDim.x + threadIdx.x;
  int n = (int)(tid >> 4), f = (int)(tid & 15);
  if (n >= Nn) return;
  float v = acc[(size_t)n * C2 + f] + b2[f];
  out2[(size_t)n * C2 + f] = v;
  if (f < 8) {
    float sp = acc[(size_t)n * C2 + 8 + f] + b2[8 + f];
    float sd = softplusf(sp) + 1e-10f;
    ixz2[(size_t)n * 8 + f] = -__logf(sd) + 0.5f * (sd * sd + v * v - 1.f);
  }
}

// ---------------------------------------------------------------------------
extern "C" void kernel_launch(void* const* d_in, const int* in_sizes, int n_in,
                              void* d_out, int out_size, void* d_ws, size_t ws_size,
                              hipStream_t stream) {
  const float* x   = (const float*)d_in[0];
  const int*   ei  = (const int*)  d_in[1];
  const float* W1  = (const float*)d_in[2];
  const float* as1 = (const float*)d_in[3];
  const float* ad1 = (const float*)d_in[4];
  const float* b1  = (const float*)d_in[5];
  const float* W2  = (const float*)d_in[6];
  const float* as2 = (const float*)d_in[7];
  const float* ad2 = (const float*)d_in[8];
  const float* b2  = (const float*)d_in[9];

  const int N    = in_sizes[0] / F_IN;
  const int E    = in_sizes[1] / 2;
  const int Etot = E + N;
  float* out = (float*)d_out;

  // workspace layout (floats)
  const size_t N64 = (size_t)N * C1, N16 = (size_t)N * C2;
  float* ws    = (float*)d_ws;
  float* h1    = ws;                 // N*64 (reused as elu(out1) after layer 1)
  float* out1a = h1 + N64;           // N*64
  float* s1    = out1a + N64;        // N
  float* d1    = s1 + N;             // N
  float* denom = d1 + N;             // N
  int*   amax  = (int*)(denom + N);  // N
  float* ewb   = (float*)(amax + N); // Etot
  float* h2    = ewb + Etot;         // N*16
  float* out2a = h2 + N16;           // N*16

  float* ixz1 = out + N16;             // d_out: [out2 | ixz1 | ixz2]
  float* ixz2 = out + (size_t)N * 48;

  const int tiles  = (N + 15) / 16;
  const int gblk   = (tiles + 7) / 8;
  const int eblk   = (Etot + 255) / 256;

  // ---- layer 1 ----
  init_accum<<<(int)((N64 + 255) / 256), 256, 0, stream>>>(out1a, N64, amax, denom, N);
  gemm1_wmma<<<gblk, 256, 0, stream>>>(x, W1, h1, N);
  node_dots<C1><<<(N + 255) / 256, 256, 0, stream>>>(h1, as1, ad1, s1, d1, N);
  edge_max_kernel<<<eblk, 256, 0, stream>>>(ei, E, Etot, s1, d1, ewb, amax);
  edge_sum_kernel<<<eblk, 256, 0, stream>>>(ei, E, Etot, ewb, amax, denom);
  {
    long long tot = (long long)Etot * (C1 / 4);
    edge_agg_kernel<C1><<<(int)((tot + 255) / 256), 256, 0, stream>>>(ei, E, Etot, ewb, denom, h1, out1a);
  }
  finish1_kernel<<<(int)((N64 + 255) / 256), 256, 0, stream>>>(out1a, b1, h1, ixz1, N);

  // ---- layer 2 ----  (h1 now holds elu(out1))
  init_accum<<<(int)((N16 + 255) / 256), 256, 0, stream>>>(out2a, N16, amax, denom, N);
  gemm2_wmma<<<gblk, 256, 0, stream>>>(h1, W2, h2, N);
  node_dots<C2><<<(N + 255) / 256, 256, 0, stream>>>(h2, as2, ad2, s1, d1, N);
  edge_max_kernel<<<eblk, 256, 0, stream>>>(ei, E, Etot, s1, d1, ewb, amax);
  edge_sum_kernel<<<eblk, 256, 0, stream>>>(ei, E, Etot, ewb, amax, denom);
  {
    long long tot = (long long)Etot * (C2 / 4);
    edge_agg_kernel<C2><<<(int)((tot + 255) / 256), 256, 0, stream>>>(ei, E, Etot, ewb, denom, h2, out2a);
  }
  finish2_kernel<<<(int)((N16 + 255) / 256), 256, 0, stream>>>(out2a, b2, out, ixz2, N);
}